// GraphTransformerEncoder_47639777247704
// MI455X (gfx1250) — compile-verified
//
#include <hip/hip_runtime.h>

#define NN   50000
#define EE   800000
#define HIDC 128
#define GG   64
#define LATC 64

typedef float v2f __attribute__((ext_vector_type(2)));
typedef float v8f __attribute__((ext_vector_type(8)));

// ---- monotone uint encoding of float for atomicMax-based segment max ----
__device__ __forceinline__ unsigned fkey(float f) {
  unsigned u = __float_as_uint(f);
  return (u & 0x80000000u) ? ~u : (u | 0x80000000u);
}
__device__ __forceinline__ float fdec(unsigned u) {
  return (u & 0x80000000u) ? __uint_as_float(u & 0x7fffffffu)
                           : __uint_as_float(~u);
}

// =====================================================================
// C[M, NT*16] = A[M,128] @ W[NT*16,128]^T + bias
//  - W staged into LDS via CDNA5 async global->LDS (ASYNCcnt), shared by
//    all 8 waves of the block.
//  - One wave per 16-row tile, NT 16-col accumulator tiles (v8f each).
//  - K=128 in steps of 4 with V_WMMA_F32_16X16X4_F32; next k-step's A
//    (global) and B (LDS) fragments are prefetched into registers while
//    the current 8 WMMAs issue.
// Block size must be 256 (8 waves).
// =====================================================================
template <int NT>
__global__ __launch_bounds__(256)
void gemm_wmma_f32(const float* __restrict__ A,
                   const float* __restrict__ W,
                   const float* __restrict__ bias,
                   float* __restrict__ C, int M) {
  __shared__ float Wlds[NT * 16 * HIDC];

  // ---- async-stage W into LDS (NT*2048 floats = NT*512 float4) ----
  {
    const float4* Wg = reinterpret_cast<const float4*>(W);
    float4*       Ws = reinterpret_cast<float4*>(Wlds);
#pragma unroll
    for (int i = 0; i < NT * 2; ++i) {
      const int idx = (int)threadIdx.x + i * 256;
      unsigned loff = (unsigned)(uintptr_t)(Ws + idx);   // low 32 bits = LDS offset
      const float4* g = Wg + idx;
      asm volatile("global_load_async_to_lds_b128 %0, %1, off"
                   :: "v"(loff), "v"(g) : "memory");
    }
    asm volatile("s_wait_asynccnt 0" ::: "memory");
    __syncthreads();   // all waves' async stores visible block-wide
  }

  const int lane  = threadIdx.x & 31;
  const int wave  = (int)((blockIdx.x * blockDim.x + threadIdx.x) >> 5);
  const int m0    = wave * 16;
  const bool active = (m0 < M);               // wave-uniform

  if (active) {
    const int r     = lane & 15;
    const int khalf = (lane >> 4) << 1;       // 0 or 2

    v8f acc[NT];
#pragma unroll
    for (int nt = 0; nt < NT; ++nt)
      acc[nt] = (v8f){0.f, 0.f, 0.f, 0.f, 0.f, 0.f, 0.f, 0.f};

    const float* arow = A + (size_t)(m0 + r) * HIDC + khalf;
    const float* brow = Wlds + (size_t)r * HIDC + khalf;

    // prologue: fragments for k0 = 0
    v2f a = *reinterpret_cast<const v2f*>(arow);
    v2f b[NT];
#pragma unroll
    for (int nt = 0; nt < NT; ++nt)
      b[nt] = *reinterpret_cast<const v2f*>(brow + nt * 16 * HIDC);

#pragma unroll 2
    for (int k0 = 0; k0 < HIDC - 4; k0 += 4) {
      // prefetch k0+4 fragments while current WMMAs issue
      v2f an = *reinterpret_cast<const v2f*>(arow + k0 + 4);
      v2f bn[NT];
#pragma unroll
      for (int nt = 0; nt < NT; ++nt)
        bn[nt] = *reinterpret_cast<const v2f*>(brow + nt * 16 * HIDC + k0 + 4);
#pragma unroll
      for (int nt = 0; nt < NT; ++nt)
        acc[nt] = __builtin_amdgcn_wmma_f32_16x16x4_f32(
            false, a, false, b[nt], (short)0, acc[nt], false, false);
      a = an;
#pragma unroll
      for (int nt = 0; nt < NT; ++nt) b[nt] = bn[nt];
    }
    // epilogue k-step
#pragma unroll
    for (int nt = 0; nt < NT; ++nt)
      acc[nt] = __builtin_amdgcn_wmma_f32_16x16x4_f32(
          false, a, false, b[nt], (short)0, acc[nt], false, false);

    // C/D layout: VGPR j -> row m0 + j + 8*(lane>=16), col = nt*16 + (lane&15)
    const int col_lo = lane & 15;
    const int rbase  = (lane >> 4) * 8;
    const int ldc    = NT * 16;
#pragma unroll
    for (int nt = 0; nt < NT; ++nt) {
      const int col = nt * 16 + col_lo;
      const float bv = bias[col];
#pragma unroll
      for (int j = 0; j < 8; ++j)
        C[(size_t)(m0 + rbase + j) * ldc + col] = acc[nt][j] + bv;
    }
  }
}

// ---- init per-node segment-softmax state: m = enc(-inf), s = 0 ----
__global__ void init_ms(unsigned* __restrict__ m, float* __restrict__ s, int n4) {
  int i = blockIdx.x * blockDim.x + threadIdx.x;
  if (i >= n4) return;
  m[i] = 0x007FFFFFu;   // fkey(-inf)
  s[i] = 0.f;
}

// ---- pass 1: alpha[e,h] = scale * dot(q[dst,h,:], k[src,h,:]); atomicMax m ----
__global__ void edge_alpha_max(const float* __restrict__ q,
                               const float* __restrict__ k,
                               const int* __restrict__ ei,
                               float* __restrict__ alpha,
                               unsigned* __restrict__ m,
                               float scale) {
  const int lane = threadIdx.x & 31;
  const int e    = (int)((blockIdx.x * blockDim.x + threadIdx.x) >> 5);
  if (e >= EE) return;
  const int src = ei[e];
  const int dst = ei[EE + e];
  float4 qq = reinterpret_cast<const float4*>(q + (size_t)dst * HIDC)[lane];
  float4 kk = reinterpret_cast<const float4*>(k + (size_t)src * HIDC)[lane];
  float p = qq.x * kk.x + qq.y * kk.y + qq.z * kk.z + qq.w * kk.w;
  p += __shfl_xor(p, 1, 32);
  p += __shfl_xor(p, 2, 32);
  p += __shfl_xor(p, 4, 32);                 // sum over the 8 lanes of a head
  if ((lane & 7) == 0) {
    const int hh = lane >> 3;
    const float av = p * scale;
    alpha[(size_t)e * 4 + hh] = av;
    atomicMax(&m[(size_t)dst * 4 + hh], fkey(av));
  }
}

// ---- pass 2: alpha = exp(alpha - m[dst]); s[dst] += alpha ----
__global__ void edge_exp_sum(const int* __restrict__ ei,
                             const unsigned* __restrict__ m,
                             float* __restrict__ alpha,
                             float* __restrict__ s) {
  const int i = blockIdx.x * blockDim.x + threadIdx.x;
  if (i >= EE * 4) return;
  const int e  = i >> 2;
  const int hh = i & 3;
  const int dst = ei[EE + e];
  const float ex = __expf(alpha[i] - fdec(m[(size_t)dst * 4 + hh]));
  alpha[i] = ex;
  unsafeAtomicAdd(&s[(size_t)dst * 4 + hh], ex);
}

// ---- pass 3: out[dst] += (alpha / (s[dst]+eps)) * v[src]  (out holds skip) ----
__global__ void edge_scatter(const float* __restrict__ v,
                             const int* __restrict__ ei,
                             const float* __restrict__ alpha,
                             const float* __restrict__ s,
                             float* __restrict__ out) {
  const int lane = threadIdx.x & 31;
  const int e    = (int)((blockIdx.x * blockDim.x + threadIdx.x) >> 5);
  if (e >= EE) return;
  const int src = ei[e];
  const int dst = ei[EE + e];
  const int hh  = lane >> 3;
  const float coef = alpha[(size_t)e * 4 + hh] /
                     (s[(size_t)dst * 4 + hh] + 1e-16f);
  float4 vv = reinterpret_cast<const float4*>(v + (size_t)src * HIDC)[lane];
  float* o = out + (size_t)dst * HIDC + lane * 4;
  unsafeAtomicAdd(o + 0, coef * vv.x);
  unsafeAtomicAdd(o + 1, coef * vv.y);
  unsafeAtomicAdd(o + 2, coef * vv.z);
  unsafeAtomicAdd(o + 3, coef * vv.w);
}

// ---- h = relu(out) ----
__global__ void relu_copy(const float* __restrict__ out, float* __restrict__ h, int n) {
  int i = blockIdx.x * blockDim.x + threadIdx.x;
  if (i < n) h[i] = fmaxf(out[i], 0.f);
}

__global__ void zero_buf(float* __restrict__ p, int n) {
  int i = blockIdx.x * blockDim.x + threadIdx.x;
  if (i < n) p[i] = 0.f;
}

// ---- pooled[batch[n]] += h[n]  (global_add_pool) ----
__global__ void pool_add(const float* __restrict__ h, const int* __restrict__ batch,
                         float* __restrict__ pooled) {
  int i = blockIdx.x * blockDim.x + threadIdx.x;
  if (i >= NN * HIDC) return;
  const int n = i >> 7;
  const int c = i & 127;
  unsafeAtomicAdd(&pooled[(size_t)batch[n] * HIDC + c], h[i]);
}

extern "C" void kernel_launch(void* const* d_in, const int* in_sizes, int n_in,
                              void* d_out, int out_size, void* d_ws, size_t ws_size,
                              hipStream_t stream) {
  const float* x    = (const float*)d_in[0];
  const int*   ei   = (const int*)d_in[1];
  const int*   batch= (const int*)d_in[2];
  const float* Win  = (const float*)d_in[3];
  const float* bin  = (const float*)d_in[4];
  const float* Wq   = (const float*)d_in[5];
  const float* bq   = (const float*)d_in[6];
  const float* Wk   = (const float*)d_in[7];
  const float* bk   = (const float*)d_in[8];
  const float* Wv   = (const float*)d_in[9];
  const float* bv   = (const float*)d_in[10];
  const float* Wsk  = (const float*)d_in[11];
  const float* bsk  = (const float*)d_in[12];
  const float* Wfc  = (const float*)d_in[13];
  const float* bfc  = (const float*)d_in[14];

  // workspace layout (floats): everything fits well inside the 192MB L2
  float*    h      = (float*)d_ws;
  float*    q      = h   + (size_t)NN * HIDC;
  float*    kbuf   = q   + (size_t)NN * HIDC;
  float*    vbuf   = kbuf+ (size_t)NN * HIDC;
  float*    out    = vbuf+ (size_t)NN * HIDC;
  float*    alpha  = out + (size_t)NN * HIDC;         // E*4
  float*    s      = alpha + (size_t)EE * 4;          // N*4
  unsigned* m      = (unsigned*)(s + (size_t)NN * 4); // N*4
  float*    pooled = (float*)(m + (size_t)NN * 4);    // G*128

  const int TB = 256;
  const int waves_per_blk = TB / 32;
  const int gemm_grid  = (NN / 16 + waves_per_blk - 1) / waves_per_blk; // 3125 tiles
  const int edge_grid  = (EE + waves_per_blk - 1) / waves_per_blk;      // wave/edge
  const int eh_grid    = (EE * 4 + TB - 1) / TB;
  const int nh_grid    = (NN * HIDC + TB - 1) / TB;
  const int n4_grid    = (NN * 4 + TB - 1) / TB;
  const float scale    = 0.17677669529663687f;        // 1/sqrt(32)

  // h = x @ Win^T + bin
  gemm_wmma_f32<8><<<gemm_grid, TB, 0, stream>>>(x, Win, bin, h, NN);

  for (int l = 0; l < 3; ++l) {
    const float* Wql = Wq  + (size_t)l * HIDC * HIDC;
    const float* Wkl = Wk  + (size_t)l * HIDC * HIDC;
    const float* Wvl = Wv  + (size_t)l * HIDC * HIDC;
    const float* Wsl = Wsk + (size_t)l * HIDC * HIDC;
    const float* bql = bq  + (size_t)l * HIDC;
    const float* bkl = bk  + (size_t)l * HIDC;
    const float* bvl = bv  + (size_t)l * HIDC;
    const float* bsl = bsk + (size_t)l * HIDC;

    gemm_wmma_f32<8><<<gemm_grid, TB, 0, stream>>>(h, Wql, bql, q,    NN);
    gemm_wmma_f32<8><<<gemm_grid, TB, 0, stream>>>(h, Wkl, bkl, kbuf, NN);
    gemm_wmma_f32<8><<<gemm_grid, TB, 0, stream>>>(h, Wvl, bvl, vbuf, NN);
    // skip connection written straight into the scatter accumulator
    gemm_wmma_f32<8><<<gemm_grid, TB, 0, stream>>>(h, Wsl, bsl, out,  NN);

    init_ms<<<n4_grid, TB, 0, stream>>>(m, s, NN * 4);
    edge_alpha_max<<<edge_grid, TB, 0, stream>>>(q, kbuf, ei, alpha, m, scale);
    edge_exp_sum <<<eh_grid,   TB, 0, stream>>>(ei, m, alpha, s);
    edge_scatter <<<edge_grid, TB, 0, stream>>>(vbuf, ei, alpha, s, out);
    relu_copy    <<<nh_grid,   TB, 0, stream>>>(out, h, NN * HIDC);
  }

  zero_buf<<<(GG * HIDC + TB - 1) / TB, TB, 0, stream>>>(pooled, GG * HIDC);
  pool_add<<<nh_grid, TB, 0, stream>>>(h, batch, pooled);

  // d_out[64,64] = pooled[64,128] @ Wfc[64,128]^T + bfc
  gemm_wmma_f32<4><<<1, TB, 0, stream>>>(pooled, Wfc, bfc, (float*)d_out, GG);
}